// RecurrentConvolutionalLayer_25821343383661
// MI455X (gfx1250) — compile-verified
//
#include <hip/hip_runtime.h>
#include <hip/hip_bf16.h>
#include <cstdint>

typedef float v2f __attribute__((ext_vector_type(2)));
typedef float v4f __attribute__((ext_vector_type(4)));
typedef float v8f __attribute__((ext_vector_type(8)));

#define RC_B    32
#define RC_H    112
#define RC_W    112
#define RC_CIN  64
#define RC_F    128
#define RC_NPIX (RC_B * RC_H * RC_W)   /* 401408 */
#define RC_ALPHA 0.001f
#define RC_BETA  0.75f

#define TAP_ELEMS (RC_CIN * RC_F)      /* 8192 floats per 3x3 tap   */
#define TAP_BYTES (TAP_ELEMS * 4)      /* 32768 bytes per tap slice */

// CDNA5 async global->LDS copy, 16B per lane, ASYNCcnt-tracked (no VGPR round trip).
// VGLOBAL encoding: vdst = LDS byte-offset VGPR, vaddr = 32-bit offset, saddr = base.
__device__ __forceinline__ void async_copy_b128(unsigned lds_off, unsigned goff,
                                                const void* base)
{
    asm volatile("global_load_async_to_lds_b128 %0, %1, %2"
                 :: "v"(lds_off), "v"(goff), "s"(base)
                 : "memory");
}
__device__ __forceinline__ void wait_asynccnt0()
{
    asm volatile("s_wait_asynccnt 0x0" ::: "memory");
}

// ---------------------------------------------------------------------------
// One-time weight reorder into B-fragment-native layout:
//   wk2[tap][k/2][n][k&1]  <-  wk[tap][k][n]
// so a 16x16x4 B fragment (w[k0][n], w[k0+1][n]) is one contiguous 8B LDS load.
// ---------------------------------------------------------------------------
__launch_bounds__(256)
__global__ void rcl_reorder_w(const float* __restrict__ wk,
                              float* __restrict__ wk2)
{
    const int idx = blockIdx.x * 256 + threadIdx.x;      // 9*64*128 = 73728
    const int tap = idx / TAP_ELEMS;
    const int r   = idx % TAP_ELEMS;
    const int k   = r / RC_F;
    const int n   = r % RC_F;
    wk2[tap * TAP_ELEMS + (k >> 1) * (RC_F * 2) + n * 2 + (k & 1)] = wk[idx];
}

// ---------------------------------------------------------------------------
// Forward conv as implicit GEMM on V_WMMA_F32_16X16X4_F32.
// Block = 256 threads = 8 waves; block tile = 64(M) x 128(N); wave tile = 16 x 64.
// Weights (pre-interleaved) double-buffered in LDS via async-to-LDS copies:
// while computing tap i from buffer i&1, the async engine fills buffer (i+1)&1.
// ---------------------------------------------------------------------------
__launch_bounds__(256)
__global__ void rcl_conv_fwd(const float* __restrict__ x,
                             const float* __restrict__ wk2,
                             float* __restrict__ out_fwd)
{
    __shared__ float lds_w[2][TAP_ELEMS];   // 2 x 32 KB tap slices

    const int tid    = threadIdx.x;
    const int wave   = tid >> 5;
    const int lane   = tid & 31;
    const int laneN  = lane & 15;              // column within 16-wide tile
    const int laneHi = lane >> 4;              // K-pair select (A/B frag layout)

    const int m_base = blockIdx.x * 64 + (wave & 3) * 16;
    const int n_base = (wave >> 2) * 64;

    // This lane's A-matrix row (output pixel); A 16x4 layout: M = lane%16 for
    // both half-waves, laneHi selects K=0,1 vs K=2,3.
    const int m  = m_base + laneN;
    const int b  = m / (RC_H * RC_W);
    const int hw = m % (RC_H * RC_W);
    const int ph = hw / RC_W;
    const int pw = hw % RC_W;

    const unsigned lds0 = (unsigned)(uintptr_t)(&lds_w[0][0]);

    // Stage one tap's 32KB slice: 256 threads x 8 x 16B async copies.
    auto stage_tap = [&](int tap, int buf) {
        const unsigned lbase = lds0 + (unsigned)buf * TAP_BYTES + (unsigned)tid * 16u;
        const unsigned gbase = (unsigned)tap * TAP_BYTES + (unsigned)tid * 16u;
        #pragma unroll
        for (int i = 0; i < 8; ++i)
            async_copy_b128(lbase + i * 4096u, gbase + i * 4096u, wk2);
    };

    v8f acc[4] = {};

    stage_tap(0, 0);
    wait_asynccnt0();
    __syncthreads();

    for (int tap = 0; tap < 9; ++tap) {
        if (tap < 8) stage_tap(tap + 1, (tap + 1) & 1);   // prefetch next tap

        const int  kh    = tap / 3, kw = tap % 3;
        const int  hh    = ph + kh - 1;
        const int  ww    = pw + kw - 1;
        const bool valid = ((unsigned)hh < RC_H) && ((unsigned)ww < RC_W);
        const float vm   = valid ? 1.0f : 0.0f;           // zero-pad mask
        const int  hc    = hh < 0 ? 0 : (hh > RC_H - 1 ? RC_H - 1 : hh);
        const int  wc    = ww < 0 ? 0 : (ww > RC_W - 1 ? RC_W - 1 : ww);
        // Always-in-bounds address; loads issue unpredicated, mask applied in VALU.
        const float* xa  = x + ((size_t)(b * RC_H + hc) * RC_W + wc) * RC_CIN
                             + laneHi * 2;
        const float* wb  = &lds_w[tap & 1][0];

        #pragma unroll 4
        for (int c0 = 0; c0 < RC_CIN; c0 += 4) {
            v2f a;
            a.x = xa[c0]     * vm;
            a.y = xa[c0 + 1] * vm;

            // Interleaved layout: pair p = c0/2 + laneHi holds (w[k0][n], w[k0+1][n])
            // contiguously at [p*256 + n*2] -> one aligned ds_load_b64 per fragment.
            const int prow = ((c0 >> 1) + laneHi) * (RC_F * 2);
            #pragma unroll
            for (int t = 0; t < 4; ++t) {
                const int n = n_base + t * 16 + laneN;
                const v2f bf = *(const v2f*)(wb + prow + n * 2);
                acc[t] = __builtin_amdgcn_wmma_f32_16x16x4_f32(
                    false, a, false, bf, (short)0, acc[t], false, false);
            }
        }

        wait_asynccnt0();      // this wave's next-tap copies landed
        __syncthreads();       // everyone done reading buf tap&1 / writing next
    }

    // C layout: VGPR j -> M = j + laneHi*8, N = lane%16.
    #pragma unroll
    for (int t = 0; t < 4; ++t) {
        const int n = n_base + t * 16 + laneN;
        #pragma unroll
        for (int j = 0; j < 8; ++j) {
            const int mm = m_base + j + laneHi * 8;
            out_fwd[(size_t)mm * RC_F + n] = acc[t][j];
        }
    }
}

// ---------------------------------------------------------------------------
// One recurrent iteration, fully fused:
//   out = out_fwd + (has_rec ? dwconv3x3(state_in, rk) : 0) + bias
//   if do_lrn: dst = relu(out) / (1 + alpha * window17_sum(relu^2))^beta
//   else:      dst = out                      (final iteration -> d_out)
// One wave per pixel; each lane owns 4 consecutive channels (float4).
// ---------------------------------------------------------------------------
__launch_bounds__(256)
__global__ void rcl_iterate(const float* __restrict__ out_fwd,
                            const float* __restrict__ state_in,
                            const float* __restrict__ rk,      // 9 floats
                            const float* __restrict__ bias,
                            float* __restrict__ dst,
                            int has_rec, int do_lrn)
{
    __shared__ float sq[8][RC_F + 8];

    const int    wave = threadIdx.x >> 5;
    const int    lane = threadIdx.x & 31;
    const size_t pix  = (size_t)blockIdx.x * 8 + wave;
    const int    b    = (int)(pix / (RC_H * RC_W));
    const int    hw   = (int)(pix % (RC_H * RC_W));
    const int    ph   = hw / RC_W, pw = hw % RC_W;
    const int    c4   = lane * 4;

    v4f v = *(const v4f*)(out_fwd + pix * RC_F + c4);
    v += *(const v4f*)(bias + c4);

    if (has_rec) {
        v4f rec = {};
        #pragma unroll
        for (int tap = 0; tap < 9; ++tap) {
            const int hh = ph + tap / 3 - 1;
            const int ww = pw + tap % 3 - 1;
            if (((unsigned)hh < RC_H) && ((unsigned)ww < RC_W)) {
                const float k = rk[tap];
                const v4f   s = *(const v4f*)(state_in +
                    ((size_t)(b * RC_H + hh) * RC_W + ww) * RC_F + c4);
                rec += k * s;
            }
        }
        v += rec;
    }

    if (!do_lrn) {                           // uniform across the whole grid
        *(v4f*)(dst + pix * RC_F + c4) = v;
        return;
    }

    v4f r;
    #pragma unroll
    for (int j = 0; j < 4; ++j) {
        r[j] = fmaxf(v[j], 0.0f);
        sq[wave][c4 + j] = r[j] * r[j];
    }
    __syncthreads();

    v4f st;
    #pragma unroll
    for (int j = 0; j < 4; ++j) {
        const int c  = c4 + j;
        const int lo = (c - 8 < 0) ? 0 : c - 8;
        const int hi = (c + 8 > RC_F - 1) ? RC_F - 1 : c + 8;
        float s = 0.0f;
        for (int cc = lo; cc <= hi; ++cc) s += sq[wave][cc];
        st[j] = r[j] * __powf(1.0f + RC_ALPHA * s, -RC_BETA);
    }
    *(v4f*)(dst + pix * RC_F + c4) = st;
}

// ---------------------------------------------------------------------------
extern "C" void kernel_launch(void* const* d_in, const int* in_sizes, int n_in,
                              void* d_out, int out_size, void* d_ws, size_t ws_size,
                              hipStream_t stream)
{
    const float* x    = (const float*)d_in[0];   // (32,112,112,64)
    const float* wk   = (const float*)d_in[1];   // (3,3,64,128)
    const float* rk   = (const float*)d_in[2];   // (3,3,1,1) -> 9 floats
    const float* bias = (const float*)d_in[3];   // (128,)
    float* out = (float*)d_out;                  // (32,112,112,128)

    const size_t plane = (size_t)RC_NPIX * RC_F;
    float* out_fwd = (float*)d_ws;
    float* state_a = out_fwd + plane;
    float* state_b = state_a + plane;
    float* wk2     = state_b + plane;            // 9*8192 floats = 288 KB

    dim3 blk(256);

    // Stage 0: one-time weight interleave (B-fragment-native layout)
    rcl_reorder_w<<<(9 * TAP_ELEMS) / 256, blk, 0, stream>>>(wk, wk2);

    // Stage 1: forward conv via WMMA implicit GEMM (async-LDS weight pipeline)
    rcl_conv_fwd<<<RC_NPIX / 64, blk, 0, stream>>>(x, wk2, out_fwd);

    // Iter 1: state==0 -> no recurrent term; state_a = lrn(relu(out_fwd+bias))
    rcl_iterate<<<RC_NPIX / 8, blk, 0, stream>>>(out_fwd, nullptr, rk, bias,
                                                 state_a, /*has_rec=*/0, /*do_lrn=*/1);
    // Iter 2: state_a -> state_b
    rcl_iterate<<<RC_NPIX / 8, blk, 0, stream>>>(out_fwd, state_a, rk, bias,
                                                 state_b, 1, 1);
    // Iter 3: state_b -> final output (no LRN)
    rcl_iterate<<<RC_NPIX / 8, blk, 0, stream>>>(out_fwd, state_b, rk, bias,
                                                 out, 1, 0);
}